// EvolveGCNNet_27573690040589
// MI455X (gfx1250) — compile-verified
//
#include <hip/hip_runtime.h>

// ---------------------------------------------------------------------------
// EvolveGCN-H forward, fully fused into ONE persistent workgroup on one WGP.
// All intermediates live in LDS (CDNA5: 320KB/WGP). All GEMMs use the
// full-precision fp32 WMMA V_WMMA_F32_16X16X4_F32 (gfx1250-only path).
// Inner-loop loads are unconditional (clamped indices); zero-padded A columns
// annihilate any clamped-garbage B values, so no exec-mask branch diamonds.
// ---------------------------------------------------------------------------

typedef __attribute__((ext_vector_type(2))) float v2f;
typedef __attribute__((ext_vector_type(8))) float v8f;

static constexpr int kN   = 129;   // nodes == in_channels
static constexpr int kIN  = 129;
static constexpr int kOUT = 64;
static constexpr int kE   = 4096;
static constexpr int kLD  = 132;   // padded K / row stride (mult of 4)
static constexpr int kRP  = 144;   // padded rows (9 * 16 M-tiles)
static constexpr int kKS  = 33;    // 132 / 4 k-steps per GEMM
static constexpr int kMT  = 9;     // M tiles (144/16)

static __device__ __forceinline__ v8f wmma4(v2f a, v2f b, v8f c) {
  // D = A(16x4 f32) * B(4x16 f32) + C(16x16 f32); wave32, full fp32.
  return __builtin_amdgcn_wmma_f32_16x16x4_f32(
      /*neg_a=*/false, a, /*neg_b=*/false, b,
      /*c_mod=*/(short)0, c, /*reuse_a=*/false, /*reuse_b=*/false);
}

static __device__ __forceinline__ float sigm(float x) {
  return 1.0f / (1.0f + __expf(-x));
}

__launch_bounds__(512, 1)
__global__ void evolve_gcn_fused(const float* __restrict__ x,
                                 const int*   __restrict__ eidx,   // [2][E]
                                 const float* __restrict__ ew,     // [E]
                                 const float* __restrict__ pool_p, // [129]
                                 const float* __restrict__ W0,     // [129,129]
                                 const float* __restrict__ w_ih,   // [387,129]
                                 const float* __restrict__ w_hh,   // [387,129]
                                 const float* __restrict__ b_ih,   // [387]
                                 const float* __restrict__ b_hh,   // [387]
                                 const float* __restrict__ conv_b, // [129]
                                 const float* __restrict__ lin_w,  // [64,129]
                                 const float* __restrict__ lin_b,  // [64]
                                 float*       __restrict__ out)    // [129,64]
{
  // --- LDS panels (4 * 144*132*4B = 297KB + ~2KB small arrays, < 320KB) ---
  __shared__ float s_x [kRP * kLD];   // x        (zero padded)
  __shared__ float s_xt[kRP * kLD];   // x_tilde  -> later reused as s_out
  __shared__ float s_h [kRP * kLD];   // W0       -> later reused as xw
  __shared__ float s_w [kRP * kLD];   // evolved GCN weight W
  __shared__ float s_score[kN];
  __shared__ int   s_perm [kN];
  __shared__ float s_dis  [kN];
  __shared__ float s_deg  [kN];

  const int tid    = threadIdx.x;
  const int nthr   = blockDim.x;       // 512
  const int lane   = tid & 31;
  const int wave   = tid >> 5;
  const int nwav   = nthr >> 5;        // 16 waves
  const int m16    = lane & 15;        // row (A) / col (B,C) lane index
  const int hh     = lane >> 4;        // wave half -> k/row-half selector

  // ---- Phase 0: zero-fill LDS (padding rows/cols must be exactly 0) ----
  for (int i = tid; i < kRP * kLD; i += nthr) {
    s_x[i] = 0.f; s_xt[i] = 0.f; s_h[i] = 0.f; s_w[i] = 0.f;
  }
  for (int i = tid; i < kN; i += nthr) s_deg[i] = 0.f;
  __syncthreads();

  // ---- Phase 1: stage x, W0 into LDS; accumulate in-degrees ----
  for (int i = tid; i < kN * kIN; i += nthr) {
    int r = i / kIN, c = i % kIN;
    s_x[r * kLD + c] = x[i];
    s_h[r * kLD + c] = W0[i];
  }
  for (int e = tid; e < kE; e += nthr) {
    atomicAdd(&s_deg[eidx[kE + e]], ew[e]);        // col-indexed degree
  }
  for (int i = tid; i < kN; i += nthr) atomicAdd(&s_deg[i], 1.0f); // self loop
  __syncthreads();

  // ---- Phase 2: TopK scores + D^-1/2 ----
  for (int i = tid; i < kN; i += nthr) {
    float pn = 0.f, d = 0.f;
    for (int k = 0; k < kIN; ++k) { float pv = pool_p[k]; pn += pv * pv; }
    for (int k = 0; k < kIN; ++k) d += s_x[i * kLD + k] * pool_p[k];
    s_score[i] = tanhf(d * rsqrtf(pn));
    float dg = s_deg[i];
    s_dis[i] = dg > 0.f ? rsqrtf(fmaxf(dg, 1e-12f)) : 0.f;
  }
  __syncthreads();

  // ---- Phase 3: stable descending argsort via counting rank (N=129) ----
  for (int i = tid; i < kN; i += nthr) {
    float si = s_score[i];
    int rank = 0;
    for (int j = 0; j < kN; ++j) {
      float sj = s_score[j];
      rank += (sj > si) || (sj == si && j < i);
    }
    s_perm[rank] = i;
  }
  __syncthreads();

  // ---- Phase 4: x_tilde = x[perm] * score[perm] ----
  for (int i = tid; i < kN * kLD; i += nthr) {
    int r = i / kLD, c = i % kLD;
    int src = s_perm[r];
    s_xt[r * kLD + c] = s_x[src * kLD + c] * s_score[src];  // pad cols stay 0
  }
  __syncthreads();

  // ---- Phase 5: GRU step. Per wave: one 16x16 tile of W. ----
  // r = sig(xt@Wir^T + W0@Whr^T + br),  z = sig(... z ...),
  // cand = tanh((xt@Win^T + bin) + r*(W0@Whn^T + bhn)),
  // W = (1-z)*cand + z*W0. gi+gh for r,z fuse into one accumulator chain.
  // All loads unconditional: clamped indices; k>=129 garbage is annihilated
  // by zero A columns; jn>=129 output columns are dead downstream.
  for (int t = wave; t < kMT * kMT; t += nwav) {
    int mt = t / kMT, nt = t % kMT;
    int jn = nt * 16 + m16;                 // output column (0..143)
    // row bases inside [0,387): r-gate row jn<=143, z-gate 129+jn<=272 OK;
    // n-gate clamped to last row.
    const int rowr = jn * kIN;
    const int rowz = (kN + jn) * kIN;
    const int rown = min(2 * kN + jn, 3 * kN - 1) * kIN;
    float br = b_ih[min(jn, 3 * kN - 1)]          + b_hh[min(jn, 3 * kN - 1)];
    float bz = b_ih[min(kN + jn, 3 * kN - 1)]     + b_hh[min(kN + jn, 3 * kN - 1)];
    float bi = b_ih[min(2 * kN + jn, 3 * kN - 1)];
    float bh = b_hh[min(2 * kN + jn, 3 * kN - 1)];
    v8f ar, az, ai, ah;
#pragma unroll
    for (int v = 0; v < 8; ++v) { ar[v] = br; az[v] = bz; ai[v] = bi; ah[v] = bh; }

    const int arow = (mt * 16 + m16) * kLD;           // A rows < 144, padded
    for (int ks = 0; ks < kKS; ++ks) {
      int k0 = ks * 4 + 2 * hh;                       // k = 2h+v per ISA layout
      v2f axt, ah0;
      axt.x = s_xt[arow + k0]; axt.y = s_xt[arow + k0 + 1];
      ah0.x = s_h [arow + k0]; ah0.y = s_h [arow + k0 + 1];
      int i0 = min(k0,     kIN - 1);                  // shared clamped k-index
      int i1 = min(k0 + 1, kIN - 1);
      v2f bri, bzi, bni, brh, bzh, bnh;
      bri.x = w_ih[rowr + i0];  bri.y = w_ih[rowr + i1];
      bzi.x = w_ih[rowz + i0];  bzi.y = w_ih[rowz + i1];
      bni.x = w_ih[rown + i0];  bni.y = w_ih[rown + i1];
      brh.x = w_hh[rowr + i0];  brh.y = w_hh[rowr + i1];
      bzh.x = w_hh[rowz + i0];  bzh.y = w_hh[rowz + i1];
      bnh.x = w_hh[rown + i0];  bnh.y = w_hh[rown + i1];
      ar = wmma4(axt, bri, ar);   ar = wmma4(ah0, brh, ar);
      az = wmma4(axt, bzi, az);   az = wmma4(ah0, bzh, az);
      ai = wmma4(axt, bni, ai);
      ah = wmma4(ah0, bnh, ah);
    }
    if (jn < kLD) {                                    // keep stores in-row
#pragma unroll
      for (int v = 0; v < 8; ++v) {
        int row = mt * 16 + v + 8 * hh;                // C layout: m = v+8h
        float rg  = sigm(ar[v]);
        float zg  = sigm(az[v]);
        float cnd = tanhf(ai[v] + rg * ah[v]);
        float w0v = s_h[row * kLD + jn];               // W0 (pad -> 0)
        s_w[row * kLD + jn] = (1.f - zg) * cnd + zg * w0v;
      }
    }
  }
  __syncthreads();

  // ---- Phase 6: xw = x @ W  (A=s_x, B=s_w, D overwrites s_h) ----
  for (int t = wave; t < kMT * kMT; t += nwav) {
    int mt = t / kMT, nt = t % kMT;
    int jn = nt * 16 + m16;
    int jn_c = min(jn, kLD - 1);            // unconditional LDS B loads
    v8f acc = {0.f, 0.f, 0.f, 0.f, 0.f, 0.f, 0.f, 0.f};
    const int arow = (mt * 16 + m16) * kLD;
    for (int ks = 0; ks < kKS; ++ks) {
      int k0 = ks * 4 + 2 * hh;
      v2f a; a.x = s_x[arow + k0];           a.y = s_x[arow + k0 + 1];
      v2f b; b.x = s_w[k0 * kLD + jn_c];     b.y = s_w[(k0 + 1) * kLD + jn_c];
      acc = wmma4(a, b, acc);
    }
    if (jn < kLD) {
#pragma unroll
      for (int v = 0; v < 8; ++v)
        s_h[(mt * 16 + v + 8 * hh) * kLD + jn] = acc[v];
    }
  }
  __syncthreads();

  // ---- Phase 7: zero out-accumulator (reuse s_xt) ----
  float* s_out = s_xt;
  for (int i = tid; i < kRP * kLD; i += nthr) s_out[i] = 0.f;
  __syncthreads();

  // ---- Phase 8: edge scatter  out[col] += dis[row]*w*dis[col] * xw[row] ----
  for (int e = wave; e < kE + kN; e += nwav) {
    int r, c; float wgt;
    if (e < kE) { r = eidx[e]; c = eidx[kE + e]; wgt = ew[e]; }
    else        { r = e - kE;  c = r;            wgt = 1.0f; }
    float nrm = s_dis[r] * wgt * s_dis[c];
    for (int f = lane; f < kN; f += 32)
      atomicAdd(&s_out[c * kLD + f], nrm * s_h[r * kLD + f]);
  }
  __syncthreads();

  // ---- Phase 9: + conv_bias, ELU (in place; pad rows/cols stay 0) ----
  for (int i = tid; i < kN * kN; i += nthr) {
    int r = i / kN, c = i % kN;
    float v = s_out[r * kLD + c] + conv_b[c];
    s_out[r * kLD + c] = v > 0.f ? v : (__expf(v) - 1.f);
  }
  __syncthreads();

  // ---- Phase 10: final linear  out = elu(h) @ lin_w^T + lin_b ----
  for (int t = wave; t < kMT * 4; t += nwav) {       // 9 x 4 tiles (64 cols)
    int mt = t / 4, ot = t % 4;
    int oc = ot * 16 + m16;                          // < 64 always
    float lb = lin_b[oc];
    v8f acc;
#pragma unroll
    for (int v = 0; v < 8; ++v) acc[v] = lb;
    const int arow = (mt * 16 + m16) * kLD;
    const int rowo = oc * kIN;
    for (int ks = 0; ks < kKS; ++ks) {
      int k0 = ks * 4 + 2 * hh;
      v2f a; a.x = s_out[arow + k0]; a.y = s_out[arow + k0 + 1];
      int i0 = min(k0,     kIN - 1);                 // max flat idx 8255, safe
      int i1 = min(k0 + 1, kIN - 1);
      v2f b; b.x = lin_w[rowo + i0]; b.y = lin_w[rowo + i1];
      acc = wmma4(a, b, acc);
    }
#pragma unroll
    for (int v = 0; v < 8; ++v) {
      int row = mt * 16 + v + 8 * hh;
      if (row < kN) out[row * kOUT + oc] = acc[v];
    }
  }
}

extern "C" void kernel_launch(void* const* d_in, const int* in_sizes, int n_in,
                              void* d_out, int out_size, void* d_ws, size_t ws_size,
                              hipStream_t stream) {
  (void)in_sizes; (void)n_in; (void)out_size; (void)d_ws; (void)ws_size;
  const float* x       = (const float*)d_in[0];
  const int*   eidx    = (const int*)  d_in[1];
  const float* ew      = (const float*)d_in[2];
  const float* pool_p  = (const float*)d_in[3];
  const float* W0      = (const float*)d_in[4];
  const float* w_ih    = (const float*)d_in[5];
  const float* w_hh    = (const float*)d_in[6];
  const float* b_ih    = (const float*)d_in[7];
  const float* b_hh    = (const float*)d_in[8];
  const float* conv_b  = (const float*)d_in[9];
  const float* lin_w   = (const float*)d_in[10];
  const float* lin_b   = (const float*)d_in[11];
  evolve_gcn_fused<<<1, 512, 0, stream>>>(x, eidx, ew, pool_p, W0, w_ih, w_hh,
                                          b_ih, b_hh, conv_b, lin_w, lin_b,
                                          (float*)d_out);
}